// Model15_64630667870284
// MI455X (gfx1250) — compile-verified
//
#include <hip/hip_runtime.h>
#include <hip/hip_bf16.h>

#define N_NODES  100000
#define N_EDGES  3200000
#define N_GRAPHS 1000

typedef __attribute__((ext_vector_type(2))) float v2f;
typedef __attribute__((ext_vector_type(8))) float v8f;

// Branch-free tanh: e = 2^(-2*log2(e)*|x|) via one v_exp_f32, then
// (1-e)/(1+e) with raw v_rcp_f32. Saturates to +-1 for large |x|.
__device__ __forceinline__ float fast_tanh(float v) {
    const float a = __builtin_fabsf(v);
    const float e = __builtin_amdgcn_exp2f(-2.885390082f * a);
    const float y = (1.0f - e) * __builtin_amdgcn_rcpf(1.0f + e);
    return __builtin_copysignf(y, v);
}

// ---------------------------------------------------------------------------
// Kernel 0: zero workspace (x accumulator + g accumulator)
// ---------------------------------------------------------------------------
__global__ void zero_kernel(float* __restrict__ p, int n) {
    int i = blockIdx.x * blockDim.x + threadIdx.x;
    if (i < n) p[i] = 0.0f;
}

// ---------------------------------------------------------------------------
// Kernel 1: per-edge message via WMMA f32 16x16x4 (K=32 in 8 chained steps),
//           fast tanh, scatter-add into x[N,10].
// Persistent waves: one wave handles many 16-edge tiles; B/bias loaded once.
// ---------------------------------------------------------------------------
__global__ void __launch_bounds__(256)
edge_msg_kernel(const int*   __restrict__ edge_index,  // [2, E] flat
                const float* __restrict__ node_attr,   // [N, 16]
                const float* __restrict__ edge_attr,   // [E, 16]
                const float* __restrict__ W_mpl,       // [32, 10]
                const float* __restrict__ b_mpl,       // [10]
                float*       __restrict__ x)           // [N, 10] accum
{
    const int lane = threadIdx.x & 31;
    const int wave = threadIdx.x >> 5;
    const int h = lane >> 4;   // half-wave: K pair 2h,2h+1 of each K-step per ISA A/B layout
    const int r = lane & 15;   // row (edge) within tile for A; col for B/D

    // ---- Per-wave invariant setup (branchless: clamp + mask) ----
    const int   rc   = (r < 10) ? r : 0;
    const float mask = (r < 10) ? 1.0f : 0.0f;

    // B-matrix 4x16 f32 per K-step (W_mpl rows 4k..4k+3, cols padded to 16):
    //   lanes 0-15 : VGPR0 = K=4k   row, VGPR1 = K=4k+1 row (N = lane)
    //   lanes 16-31: VGPR0 = K=4k+2 row, VGPR1 = K=4k+3 row (N = lane-16)
    v2f B[8];
    #pragma unroll
    for (int k = 0; k < 8; ++k) {
        B[k].x = W_mpl[(4 * k + 2 * h    ) * 10 + rc] * mask;
        B[k].y = W_mpl[(4 * k + 2 * h + 1) * 10 + rc] * mask;
    }
    const float bias = b_mpl[rc];

    // ---- Grid-stride loop over 16-edge tiles (wave-uniform -> EXEC full) ----
    const long NTILES  = N_EDGES / 16;
    const long wstride = (long)gridDim.x * 8;
    for (long tile = (long)blockIdx.x * 8 + wave; tile < NTILES; tile += wstride) {
        const long e0 = tile * 16;
        const long e  = e0 + r;

        const int src = edge_index[e];                 // row 0: src
        const float* nrow = node_attr + (long)src * 16;
        const float* erow = edge_attr + e * 16;

        // Prefetch the streamed edge_attr well ahead.
        if (e + 4096 < (long)N_EDGES)
            __builtin_prefetch(edge_attr + (e + 4096) * 16, 0, 1);

        // A-matrix 16x4 f32 per K-step k (features 4k..4k+3):
        //   lanes 0-15 : VGPR0 = K=4k,   VGPR1 = K=4k+1
        //   lanes 16-31: VGPR0 = K=4k+2, VGPR1 = K=4k+3
        // Features 0..15 from node_attr[src], 16..31 from edge_attr[e].
        v2f A[8];
        #pragma unroll
        for (int k = 0; k < 4; ++k)
            A[k] = *(const v2f*)(nrow + 4 * k + 2 * h);
        #pragma unroll
        for (int k = 4; k < 8; ++k)
            A[k] = *(const v2f*)(erow + 4 * (k - 4) + 2 * h);

        // Chain 8 WMMAs: full K=32 fp32 dot product accumulated in C.
        v8f C = {};
        #pragma unroll
        for (int k = 0; k < 8; ++k) {
            C = __builtin_amdgcn_wmma_f32_16x16x4_f32(
                    false, A[k], false, B[k], (short)0, C, false, false);
        }

        // dst (row 1) read once per edge by its own lane, then broadcast to
        // column-lanes via shuffle (all lanes active here).
        const int dstAll = edge_index[(long)N_EDGES + e];
        int dstj[8];
        #pragma unroll
        for (int j = 0; j < 8; ++j)
            dstj[j] = __shfl(dstAll, j + 8 * h);

        // D layout: VGPR j -> (M=j, N=lane) lanes 0-15, (M=j+8, N=lane-16)
        // lanes 16-31. Column n = r (valid n < 10).
        if (r < 10) {
            #pragma unroll
            for (int j = 0; j < 8; ++j) {
                const float v = fast_tanh(C[j] + bias);
                atomicAdd(&x[(long)dstj[j] * 10 + r], v);
            }
        }
    }
}

// ---------------------------------------------------------------------------
// Kernel 2: node MLP  x->tanh(xW1+b1)->tanh(.W2+b2), then segment-sum into g
//           by graph id, with wave-level reduction when gidx is wave-uniform
//           (batch is sorted, so this is the common case).
// ---------------------------------------------------------------------------
__global__ void __launch_bounds__(256)
node_mlp_kernel(const float* __restrict__ x,      // [N, 10]
                const int*   __restrict__ batch,  // [N]
                const float* __restrict__ W1, const float* __restrict__ b1,
                const float* __restrict__ W2, const float* __restrict__ b2,
                float*       __restrict__ g)      // [G, 5] accum
{
    const int n = blockIdx.x * blockDim.x + threadIdx.x;
    if (n >= N_NODES) return;   // N_NODES % 32 == 0 -> waves fully on/off

    float h0[10];
    #pragma unroll
    for (int i = 0; i < 10; ++i) h0[i] = x[(long)n * 10 + i];

    float h1[10];
    #pragma unroll
    for (int o = 0; o < 10; ++o) {
        float s = b1[o];
        #pragma unroll
        for (int i = 0; i < 10; ++i) s += h0[i] * W1[i * 10 + o];
        h1[o] = fast_tanh(s);
    }

    float h2[5];
    #pragma unroll
    for (int o = 0; o < 5; ++o) {
        float s = b2[o];
        #pragma unroll
        for (int i = 0; i < 10; ++i) s += h1[i] * W2[i * 5 + o];
        h2[o] = fast_tanh(s);
    }

    const int gidx  = batch[n];
    const int first = __shfl(gidx, 0);
    const int lane  = threadIdx.x & 31;
    if (__all(gidx == first)) {
        // Whole wave belongs to one graph: shuffle-reduce, one atomic/lane0.
        #pragma unroll
        for (int o = 0; o < 5; ++o) {
            float v = h2[o];
            #pragma unroll
            for (int off = 16; off > 0; off >>= 1)
                v += __shfl_down(v, off);
            if (lane == 0) atomicAdd(&g[(long)first * 5 + o], v);
        }
    } else {
        #pragma unroll
        for (int o = 0; o < 5; ++o)
            atomicAdd(&g[(long)gidx * 5 + o], h2[o]);
    }
}

// ---------------------------------------------------------------------------
// Kernel 3: per-graph head  tanh(gW3+b3) @ W4 + b4 -> out[G,1]
// ---------------------------------------------------------------------------
__global__ void graph_head_kernel(const float* __restrict__ g,
                                  const float* __restrict__ W3,
                                  const float* __restrict__ b3,
                                  const float* __restrict__ W4,
                                  const float* __restrict__ b4,
                                  float* __restrict__ out)
{
    const int i = blockIdx.x * blockDim.x + threadIdx.x;
    if (i >= N_GRAPHS) return;

    float gi[5];
    #pragma unroll
    for (int k = 0; k < 5; ++k) gi[k] = g[(long)i * 5 + k];

    float t[5];
    #pragma unroll
    for (int o = 0; o < 5; ++o) {
        float s = b3[o];
        #pragma unroll
        for (int k = 0; k < 5; ++k) s += gi[k] * W3[k * 5 + o];
        t[o] = fast_tanh(s);
    }

    float s = b4[0];
    #pragma unroll
    for (int k = 0; k < 5; ++k) s += t[k] * W4[k];
    out[i] = s;
}

// ---------------------------------------------------------------------------
extern "C" void kernel_launch(void* const* d_in, const int* in_sizes, int n_in,
                              void* d_out, int out_size, void* d_ws, size_t ws_size,
                              hipStream_t stream) {
    const int*   edge_index = (const int*)  d_in[0];  // [2, E]
    const float* node_attr  = (const float*)d_in[1];  // [N, 16]
    const float* edge_attr  = (const float*)d_in[2];  // [E, 16]
    const int*   batch      = (const int*)  d_in[3];  // [N]
    const float* W_mpl      = (const float*)d_in[4];  // [32, 10]
    const float* b_mpl      = (const float*)d_in[5];  // [10]
    const float* W1         = (const float*)d_in[6];  // [10, 10]
    const float* b1         = (const float*)d_in[7];  // [10]
    const float* W2         = (const float*)d_in[8];  // [10, 5]
    const float* b2         = (const float*)d_in[9];  // [5]
    const float* W3         = (const float*)d_in[10]; // [5, 5]
    const float* b3         = (const float*)d_in[11]; // [5]
    const float* W4         = (const float*)d_in[12]; // [5, 1]
    const float* b4         = (const float*)d_in[13]; // [1]
    float*       out        = (float*)d_out;          // [G, 1]

    // Workspace layout: x [N*10] floats, then g [G*5] floats.
    float* x = (float*)d_ws;
    float* g = x + (size_t)N_NODES * 10;
    const int accum_elems = N_NODES * 10 + N_GRAPHS * 5;

    // Zero accumulators.
    zero_kernel<<<(accum_elems + 255) / 256, 256, 0, stream>>>(x, accum_elems);

    // Edge messages: persistent waves, 16 edges per wave-iteration.
    // 200,000 tiles total; 2000 blocks x 8 waves -> ~12.5 tiles per wave.
    edge_msg_kernel<<<2000, 256, 0, stream>>>(
        edge_index, node_attr, edge_attr, W_mpl, b_mpl, x);

    // Node MLP + pooling.
    node_mlp_kernel<<<(N_NODES + 255) / 256, 256, 0, stream>>>(
        x, batch, W1, b1, W2, b2, g);

    // Graph head.
    graph_head_kernel<<<(N_GRAPHS + 255) / 256, 256, 0, stream>>>(
        g, W3, b3, W4, b4, out);
}